// AttentionHead_17051020165634
// MI455X (gfx1250) — compile-verified
//
#include <hip/hip_runtime.h>
#include <math.h>

// ---------------------------------------------------------------------------
// Single-head causal attention, fp32, flash-attention style, CDNA5 WMMA.
//   x [4,4096,100] fp32; Wq/Wk/Wv [100,64] fp32 -> out [4,4096,64] fp32
// All matmuls via V_WMMA_F32_16X16X4_F32 (fp32 A/B, fp32 accumulate).
//
// Attention computes S^T = K @ Q^T and O^T = V^T @ P^T so that softmax
// statistics are lane-local (one query per lane) -> minimal cross-lane VALU.
// ---------------------------------------------------------------------------

typedef __attribute__((ext_vector_type(2))) float v2f;
typedef __attribute__((ext_vector_type(8))) float v8f;

#define NB    4
#define T_SEQ 4096
#define EMB   100
#define HD    64
#define QTILES (NB * T_SEQ / 16)   // 1024 16-row tiles over the flattened B*T dim

__device__ __forceinline__ v8f wmma4(v2f a, v2f b, v8f c) {
  // 8-arg pattern: (neg_a, A, neg_b, B, c_mod, C, reuse_a, reuse_b)
  return __builtin_amdgcn_wmma_f32_16x16x4_f32(false, a, false, b, (short)0, c,
                                               false, false);
}

__device__ __forceinline__ void lds_order_barrier() {
#if __has_builtin(__builtin_amdgcn_s_wait_dscnt)
  __builtin_amdgcn_s_wait_dscnt(0);
  asm volatile("" ::: "memory");
#else
  asm volatile("s_wait_dscnt 0x0" ::: "memory");
#endif
}

// ---------------------------------------------------------------------------
// Kernel 1: q/k/v projections.  One wave handles one (16-row tile, matrix)
// pair: C(16x64) += A(16x4) * B(4x64) over 25 K-steps (EMB = 100 = 25*4).
//
// fp32 WMMA 16x16x4 lane layout assumed (mirrors ISA A-matrix 16x4 table):
//   A: lane l (l<16): M=l, {K=kk+0, K=kk+1}; lane l+16: M=l, {K=kk+2, K=kk+3}
//   B: lane n (n<16): N=n, rows {kk+0, kk+1}; lane n+16: N=n, rows {kk+2,kk+3}
//   C/D: lane n (n<16): N=n, VGPR r -> M=r; lane n+16: N=n, VGPR r -> M=r+8
// ---------------------------------------------------------------------------
__global__ __launch_bounds__(64) void proj_kernel(
    const float* __restrict__ x, const float* __restrict__ Wq,
    const float* __restrict__ Wk, const float* __restrict__ Wv,
    float* __restrict__ qkv /* [3][NB*T_SEQ][HD] */) {
  const int lane = threadIdx.x & 31;
  const int wave = threadIdx.x >> 5;
  const int half = lane >> 4;   // 0: K lo pair, 1: K hi pair
  const int idx  = lane & 15;   // A row / B,C column within the tile

  const int w    = blockIdx.x * 2 + wave;   // 0 .. 3*QTILES-1
  const int mat  = w % 3;                   // 0=q 1=k 2=v
  const int tile = w / 3;                   // 16-row tile over B*T

  const float* W = (mat == 0) ? Wq : (mat == 1) ? Wk : Wv;
  const float* xrow = x + (size_t)(tile * 16 + idx) * EMB;

  v8f acc[4];
  #pragma unroll
  for (int t = 0; t < 4; ++t) acc[t] = (v8f){0.f, 0.f, 0.f, 0.f, 0.f, 0.f, 0.f, 0.f};

  for (int kk = 0; kk < EMB; kk += 4) {
    const v2f a = *(const v2f*)(xrow + kk + 2 * half);   // 8B aligned
    #pragma unroll
    for (int t = 0; t < 4; ++t) {
      v2f b;
      b.x = W[(kk + 2 * half)     * HD + t * 16 + idx];
      b.y = W[(kk + 2 * half + 1) * HD + t * 16 + idx];
      acc[t] = wmma4(a, b, acc[t]);
    }
  }

  float* ob = qkv + (size_t)mat * (NB * T_SEQ * HD);
  #pragma unroll
  for (int t = 0; t < 4; ++t)
    #pragma unroll
    for (int r = 0; r < 8; ++r)
      ob[(size_t)(tile * 16 + r + 8 * half) * HD + t * 16 + idx] = acc[t][r];
}

// ---------------------------------------------------------------------------
// Kernel 2: flash attention, transposed tiles.  One wave owns one 16-query
// tile; lane idx = query column.  S^T = K @ Q^T puts each query's 16 key
// scores into 8 slots x 2 lane-halves -> softmax stats are (almost) lane-
// local: 7 lane-local ops + one shfl_xor(16) per reduction.
// ---------------------------------------------------------------------------
__global__ __launch_bounds__(64) void attn_kernel(
    const float* __restrict__ qkv, float* __restrict__ out) {
  __shared__ float ldsP[2][256];   // per-wave P(16x16) staging, query-major

  const int lane = threadIdx.x & 31;
  const int wave = threadIdx.x >> 5;
  const int half = lane >> 4;
  const int idx  = lane & 15;

  const int w  = blockIdx.x * 2 + wave;    // 0 .. QTILES-1
  const int b  = w / (T_SEQ / 16);
  const int qt = w % (T_SEQ / 16);

  const float* qp = qkv + (size_t)b * T_SEQ * HD;
  const float* kp = qkv + (size_t)(NB * T_SEQ * HD) + (size_t)b * T_SEQ * HD;
  const float* vp = qkv + (size_t)(2 * NB * T_SEQ * HD) + (size_t)b * T_SEQ * HD;

  // Preload Q^T B-fragments for all 16 K-steps: identical per-lane layout to
  // the A-fragments of Q (B-layout of X^T == A-layout of X).
  const float* qrow = qp + (size_t)(qt * 16 + idx) * HD;
  v2f qb[16];
  #pragma unroll
  for (int s = 0; s < 16; ++s) qb[s] = *(const v2f*)(qrow + 4 * s + 2 * half);

  // O^T accumulators: tile t covers head dims [16t,16t+16); lane = query.
  v8f o[4];
  #pragma unroll
  for (int t = 0; t < 4; ++t) o[t] = (v8f){0.f, 0.f, 0.f, 0.f, 0.f, 0.f, 0.f, 0.f};

  float mq = -INFINITY;   // running row max for this lane's query
  float lq = 0.f;         // running row sum for this lane's query

  const float scale = 0.125f;   // 64^-0.5
  float* pb = &ldsP[wave][0];
  const int qbase = qt * 16;
  const int qrow_id = qbase + idx;   // this lane's query index

  for (int j = 0; j <= qt; ++j) {
    // ---- S^T = K_tile(16x64) @ Q^T(64x16): 16 WMMA steps ------------------
    v8f s8 = (v8f){0.f, 0.f, 0.f, 0.f, 0.f, 0.f, 0.f, 0.f};
    const float* krow = kp + (size_t)(j * 16 + idx) * HD;   // A row = key idx
    #pragma unroll
    for (int s = 0; s < 16; ++s) {
      const v2f af = *(const v2f*)(krow + 4 * s + 2 * half);
      s8 = wmma4(af, qb[s], s8);
    }
    // s8[r] = score(query = qrow_id, key = j*16 + r + 8*half)

    // ---- scale + faithful mask: tril, then (==0 -> -inf) ------------------
    float p[8];
    float tmax = -INFINITY;
    #pragma unroll
    for (int r = 0; r < 8; ++r) {
      float sv = s8[r] * scale;
      const int key = j * 16 + r + 8 * half;
      const bool dead = (key > qrow_id) || (sv == 0.0f);
      p[r] = dead ? -INFINITY : sv;
      tmax = fmaxf(tmax, p[r]);
    }
    tmax = fmaxf(tmax, __shfl_xor(tmax, 16, 32));   // combine the two halves

    // ---- online softmax (lane-scalar stats) -------------------------------
    const float mnew  = fmaxf(mq, tmax);
    const float alpha = (mq == -INFINITY) ? 0.f : __expf(mq - mnew);
    float rs = 0.f;
    #pragma unroll
    for (int r = 0; r < 8; ++r) {
      const float e = (p[r] == -INFINITY) ? 0.f : __expf(p[r] - mnew);
      p[r] = e;
      rs += e;
    }
    rs += __shfl_xor(rs, 16, 32);
    lq = lq * alpha + rs;
    mq = mnew;
    #pragma unroll
    for (int t = 0; t < 4; ++t)
      #pragma unroll
      for (int r = 0; r < 8; ++r) o[t][r] *= alpha;

    // ---- stage P query-major in LDS: pb[query*16 + key] -------------------
    // lane(h,idx) slot r holds P[idx][r+8h] -> 4 contiguous b64 stores.
    #pragma unroll
    for (int e = 0; e < 4; ++e) {
      v2f st; st.x = p[2 * e]; st.y = p[2 * e + 1];
      *(v2f*)(pb + idx * 16 + 8 * half + 2 * e) = st;
    }
    lds_order_barrier();
    // B-fragments of P^T: {P[idx][4s+2h], P[idx][4s+2h+1]} -> b64 loads.
    v2f pf[4];
    #pragma unroll
    for (int s = 0; s < 4; ++s)
      pf[s] = *(const v2f*)(pb + idx * 16 + 4 * s + 2 * half);
    lds_order_barrier();

    // ---- O^T += V^T_tile(16x16) @ P^T(16x16) per head-dim tile ------------
    // A = V^T: a = {V[key=4s+2h][16t+idx], V[key=4s+2h+1][16t+idx]}
    const float* vt = vp + (size_t)j * 16 * HD;
    #pragma unroll
    for (int t = 0; t < 4; ++t) {
      #pragma unroll
      for (int s = 0; s < 4; ++s) {
        v2f vf;
        vf.x = vt[(4 * s + 2 * half)     * HD + t * 16 + idx];
        vf.y = vt[(4 * s + 2 * half + 1) * HD + t * 16 + idx];
        o[t] = wmma4(vf, pf[s], o[t]);
      }
    }
  }

  // ---- normalize (lane-scalar 1/l) and store O[query][d] ------------------
  // o[t][r] = O^T[16t + r + 8*half][query=idx]; pairs are contiguous in d.
  const float inv = 1.0f / lq;
  float* orow = out + ((size_t)b * T_SEQ + qrow_id) * HD;
  #pragma unroll
  for (int t = 0; t < 4; ++t) {
    #pragma unroll
    for (int e = 0; e < 4; ++e) {
      v2f st;
      st.x = o[t][2 * e]     * inv;
      st.y = o[t][2 * e + 1] * inv;
      *(v2f*)(orow + 16 * t + 8 * half + 2 * e) = st;
    }
  }
}

// ---------------------------------------------------------------------------
extern "C" void kernel_launch(void* const* d_in, const int* in_sizes, int n_in,
                              void* d_out, int out_size, void* d_ws, size_t ws_size,
                              hipStream_t stream) {
  const float* x  = (const float*)d_in[0];
  const float* Wq = (const float*)d_in[1];
  const float* Wk = (const float*)d_in[2];
  const float* Wv = (const float*)d_in[3];
  float* out = (float*)d_out;
  float* qkv = (float*)d_ws;   // needs 3 * 4*4096*64 * 4B = 12 MB of scratch

  // proj: 3*QTILES waves, 2 waves (64 threads) per block
  proj_kernel<<<dim3(3 * QTILES / 2), dim3(64), 0, stream>>>(x, Wq, Wk, Wv, qkv);
  // attention: QTILES waves, 2 waves per block
  attn_kernel<<<dim3(QTILES / 2), dim3(64), 0, stream>>>(qkv, out);
}